// Codebook_37306085933614
// MI455X (gfx1250) — compile-verified
//
#include <hip/hip_runtime.h>
#include <math.h>

// Problem constants (from reference)
#define N_ROWS  32768      // 8 * 16*16*16 latent vectors
#define C_DIM   256        // latent dim (WMMA K-reduction dim)
#define K_CODES 1024       // codebook size
#define SPB     4096       // spatial elems per batch (16^3)
#define ROWS_PER_WG 128    // 8 waves * 16 rows
#define EB_STRIDE 260      // padded LDS row stride (floats): banks 4*ln+c, conflict-free;
                           // 260*4 = 1040 B keeps rows 16B-aligned for b128 async writes

typedef float v2f __attribute__((ext_vector_type(2)));
typedef float v8f __attribute__((ext_vector_type(8)));
typedef int   v4i __attribute__((vector_size(4 * sizeof(int))));

// Async Global->LDS path (gfx1250); guarded so host pass / older toolchains fall back.
#if __has_builtin(__builtin_amdgcn_global_load_async_to_lds_b128) && \
    __has_builtin(__builtin_amdgcn_s_wait_asynccnt)
#define VQ_ASYNC 1
#else
#define VQ_ASYNC 0
#endif

// ---------------- workspace layout (4-byte units) ----------------
// [0,1024)     counts   (uint)
// [1024,1280)  partials (float, one per main block)
// [1280,2304)  e2       (float, ||e_k||^2)

__global__ __launch_bounds__(256) void vq_init(unsigned int* __restrict__ counts,
                                               float* __restrict__ partials) {
  const int t = threadIdx.x;
  for (int k = t; k < K_CODES; k += 256) counts[k] = 0u;
  partials[t] = 0.0f;
}

__global__ __launch_bounds__(256) void vq_e2(const float* __restrict__ emb,
                                             float* __restrict__ e2) {
  const int k = blockIdx.x * 256 + threadIdx.x;
  if (k < K_CODES) {
    const float* row = emb + (size_t)k * C_DIM;
    float s = 0.0f;
#pragma unroll 8
    for (int c = 0; c < C_DIM; ++c) s += row[c] * row[c];
    e2[k] = s;
  }
}

// Stage one 16x256 fp32 codebook tile (16 KB, contiguous in memory) into LDS.
// 256 threads: thread t = (row r = t>>4, base chunk = t&15), 4 x b128 each.
__device__ __forceinline__ void stage_tile(const float* __restrict__ emb, int j,
                                           float* lbuf, int tid) {
  const int r  = tid >> 4;        // 0..15 (codebook row within tile)
  const int cb = tid & 15;        // 0..15 (base 16B chunk within row)
  const float* g = emb + (size_t)(j * 16 + r) * C_DIM;
#pragma unroll
  for (int i = 0; i < 4; ++i) {
    const int q = cb + i * 16;    // 16B chunk index 0..63 within the 1KB row
#if VQ_ASYNC
    __builtin_amdgcn_global_load_async_to_lds_b128(
        (__attribute__((address_space(1))) v4i*)(g + q * 4),
        (__attribute__((address_space(3))) v4i*)(lbuf + r * EB_STRIDE + q * 4),
        /*offset=*/0, /*cpol=*/0);
#else
    const float4 v = *(const float4*)(g + q * 4);
    *(float4*)(lbuf + r * EB_STRIDE + q * 4) = v;
#endif
  }
}

__device__ __forceinline__ void wait_async_done() {
#if VQ_ASYNC
  __builtin_amdgcn_s_wait_asynccnt(0);
#endif
}

// Main kernel: fp32 WMMA distance matmul + argmin + gather + loss partial.
__global__ __launch_bounds__(256) void vq_main(const float* __restrict__ z,
                                               const float* __restrict__ emb,
                                               const float* __restrict__ e2,
                                               float* __restrict__ out_zq,
                                               float* __restrict__ out_idx,
                                               unsigned int* __restrict__ counts,
                                               float* __restrict__ partials) {
  __shared__ float ebuf[2][16 * EB_STRIDE];   // double-buffered B tiles (~32.5 KB)
  __shared__ int   lds_idx[ROWS_PER_WG];
  __shared__ float lds_red[256];

  const int wg   = blockIdx.x;            // 0..255
  const int row0 = wg * ROWS_PER_WG;      // global base row (all rows same batch b)
  const int b    = row0 / SPB;
  const int s0   = row0 % SPB;
  const int tid  = threadIdx.x;
  const int wave = tid >> 5;
  const int lane = tid & 31;
  const int ln   = lane & 15;             // 0..15
  const int hi   = lane >> 4;             // 0 or 1 (lane half)

  // ---------- Phase 1: A slab preload into registers ----------
  // A fragment (16x4 fp32): lanes 0-15 = rows with K={0,1}, lanes 16-31 = K={2,3}.
  // zf[n, c] = z[b, c, s], s = n % SPB -> coalesced 64B per half-wave per c.
  const float* zb = z + (size_t)b * (C_DIM * SPB) + (size_t)(s0 + wave * 16 + ln);

  v2f a[64];
#pragma unroll
  for (int cc = 0; cc < 64; ++cc) {
    const int c = cc * 4 + hi * 2;
    a[cc].x = zb[(size_t)c * SPB];
    a[cc].y = zb[(size_t)(c + 1) * SPB];
  }

  float best_v[8];
  int   best_i[8];
#pragma unroll
  for (int r = 0; r < 8; ++r) { best_v[r] = 3.4e38f; best_i[r] = 0; }

  // ---------- Phase 2: 64 codebook tiles, async double-buffered through LDS ----------
  stage_tile(emb, 0, ebuf[0], tid);

  for (int j = 0; j < K_CODES / 16; ++j) {
    const int cur = j & 1;
    wait_async_done();        // this wave's part of tile j is in LDS
    __syncthreads();          // all waves' parts visible; all reads of other buf done
    if (j + 1 < K_CODES / 16) stage_tile(emb, j + 1, ebuf[1 - cur], tid);

    // B fragment (4x16 fp32) from LDS: lane -> column n = j*16+ln, rows K = hi*2+{0,1}
    const float* brow = &ebuf[cur][ln * EB_STRIDE];
    v8f acc = {0.f, 0.f, 0.f, 0.f, 0.f, 0.f, 0.f, 0.f};
#pragma unroll
    for (int cc = 0; cc < 64; ++cc) {
      const int c = cc * 4 + hi * 2;
      v2f bf;
      bf.x = brow[c];                      // ds_load_b64, conflict-free banks
      bf.y = brow[c + 1];
      acc = __builtin_amdgcn_wmma_f32_16x16x4_f32(
          /*neg_a=*/false, a[cc], /*neg_b=*/false, bf,
          /*c_mod=*/(short)0, acc, /*reuse_a=*/false, /*reuse_b=*/false);
    }

    // distance (minus row-constant ||z||^2): d' = ||e||^2 - 2*dot
    const float e2v = e2[j * 16 + ln];
    const int   jn  = j * 16 + ln;
#pragma unroll
    for (int r = 0; r < 8; ++r) {
      const float dv = e2v - 2.0f * acc[r];
      const bool better = dv < best_v[r];            // strictly-less keeps first idx
      best_v[r] = better ? dv : best_v[r];
      best_i[r] = better ? jn : best_i[r];
    }
  }

  // ---------- Phase 3: cross-lane argmin (width 16; halves are distinct rows) ----------
#pragma unroll
  for (int r = 0; r < 8; ++r) {
    float v  = best_v[r];
    int   bi = best_i[r];
#pragma unroll
    for (int off = 1; off < 16; off <<= 1) {
      const float ov = __shfl_xor(v, off, 16);
      const int   oi = __shfl_xor(bi, off, 16);
      const bool take = (ov < v) || (ov == v && oi < bi); // first-occurrence tie-break
      v  = take ? ov : v;
      bi = take ? oi : bi;
    }
    if (ln == 0) {
      // D layout: VGPR r -> row r (lanes 0-15) / row r+8 (lanes 16-31)
      const int mloc = wave * 16 + hi * 8 + r;
      lds_idx[mloc] = bi;
      out_idx[row0 + mloc] = (float)bi;
      atomicAdd(&counts[bi], 1u);
    }
  }
  __syncthreads();

  // ---------- Phase 4: gather z_q (coalesced over spatial) + loss partial ----------
  float lsum = 0.0f;
  const int chalf = tid >> 7;             // 0 or 1
  const int sl    = tid & 127;            // local row / spatial offset
  const int qrow  = lds_idx[sl];
  const size_t zoff = (size_t)b * (C_DIM * SPB) + (size_t)(s0 + sl);
  const float* qbase = emb + (size_t)qrow * C_DIM;
#pragma unroll 4
  for (int c2 = 0; c2 < 128; ++c2) {
    const int c = c2 * 2 + chalf;
    const size_t o = zoff + (size_t)c * SPB;
    const float q  = qbase[c];
    const float zv = z[o];
    out_zq[o] = q;                        // straight-through: value == z_q
    const float d = q - zv;
    lsum += d * d;
  }

  lds_red[tid] = lsum;
  __syncthreads();
  for (int st = 128; st > 0; st >>= 1) {
    if (tid < st) lds_red[tid] += lds_red[tid + st];
    __syncthreads();
  }
  if (tid == 0) partials[wg] = lds_red[0];
}

__global__ __launch_bounds__(256) void vq_final(const unsigned int* __restrict__ counts,
                                                const float* __restrict__ partials,
                                                float* __restrict__ loss_out,
                                                float* __restrict__ perp_out) {
  __shared__ float red[256];
  const int t = threadIdx.x;

  // loss = 1.25 * mean(diff^2)   (LEGACY: 1.0 + BETA_C with equal values)
  red[t] = partials[t];
  __syncthreads();
  for (int st = 128; st > 0; st >>= 1) {
    if (t < st) red[t] += red[t + st];
    __syncthreads();
  }
  if (t == 0) *loss_out = 1.25f * red[0] / (float)(N_ROWS * C_DIM);
  __syncthreads();

  // perplexity = exp(-sum e*log(e+1e-10)), e = counts/N
  float h = 0.0f;
  for (int k = t; k < K_CODES; k += 256) {
    const float e = (float)counts[k] / (float)N_ROWS;
    h += e * logf(e + 1e-10f);
  }
  red[t] = h;
  __syncthreads();
  for (int st = 128; st > 0; st >>= 1) {
    if (t < st) red[t] += red[t + st];
    __syncthreads();
  }
  if (t == 0) *perp_out = expf(-red[0]);
}

extern "C" void kernel_launch(void* const* d_in, const int* in_sizes, int n_in,
                              void* d_out, int out_size, void* d_ws, size_t ws_size,
                              hipStream_t stream) {
  const float* z   = (const float*)d_in[0];   // [8,256,16,16,16] fp32
  const float* emb = (const float*)d_in[1];   // [1024,256] fp32

  float* out      = (float*)d_out;
  float* out_zq   = out;                         // 8388608 floats
  float* loss_out = out + 8388608;               // 1
  float* perp_out = out + 8388609;               // 1
  float* out_idx  = out + 8388610;               // 32768 (idx as float)

  unsigned int* counts   = (unsigned int*)d_ws;          // 1024
  float*        partials = (float*)d_ws + 1024;          // 256
  float*        e2       = (float*)d_ws + 1280;          // 1024

  vq_init <<<1,   256, 0, stream>>>(counts, partials);
  vq_e2   <<<4,   256, 0, stream>>>(emb, e2);
  vq_main <<<256, 256, 0, stream>>>(z, emb, e2, out_zq, out_idx, counts, partials);
  vq_final<<<1,   256, 0, stream>>>(counts, partials, loss_out, perp_out);
}